// JointEdgeSegLoss_42245298323666
// MI455X (gfx1250) — compile-verified
//
#include <hip/hip_runtime.h>
#include <hip/hip_bf16.h>

#define NCLS 19
#define NIMG 8
#define HW   (512 * 512)
#define BLK  256
#define CHUNK 4096                 // pixels per block
#define BPI  (HW / CHUNK)          // 64 blocks per image
#define NBLK (NIMG * BPI)          // 512 blocks total

// workspace layout in 32-bit words
#define OFF_CNT_SEG 0                              // uint [8*19]
#define OFF_CNT_ATT (OFF_CNT_SEG + NIMG * NCLS)    // uint [8*19]
#define OFF_POSNEG  (OFF_CNT_ATT + NIMG * NCLS)    // uint [2]
#define OFF_W_SEG   320                            // float [8*19]
#define OFF_W_ATT   (OFF_W_SEG + NIMG * NCLS)      // float [8*19]
#define OFF_PART    640                            // float [NBLK][8]

typedef __attribute__((ext_vector_type(2))) float v2f;
typedef __attribute__((ext_vector_type(8))) float v8f;

// ---------------------------------------------------------------------------
// Pass A: per-image class histograms (seg + edge-gated), global pos/neg counts
// ---------------------------------------------------------------------------
__global__ void histo_kernel(const float* __restrict__ edgein,
                             const int* __restrict__ segmask,
                             const int* __restrict__ edgemask,
                             unsigned* __restrict__ ws_u) {
    __shared__ unsigned h_seg[NCLS];
    __shared__ unsigned h_att[NCLS];
    __shared__ unsigned h_pn[2];
    const int tid = threadIdx.x;
    if (tid < NCLS) { h_seg[tid] = 0u; h_att[tid] = 0u; }
    if (tid < 2) h_pn[tid] = 0u;
    __syncthreads();

    const int n = blockIdx.y;
    const int base = n * HW + blockIdx.x * CHUNK;
    unsigned pn0 = 0, pn1 = 0;
    for (int i = 0; i < CHUNK / BLK; ++i) {
        const int p = base + i * BLK + tid;
        const int t = segmask[p];
        const float e = edgein[p];
        const int te = edgemask[p];
        const bool v = (t >= 0) && (t < NCLS);
        if (v) {
            atomicAdd(&h_seg[t], 1u);
            if (e > 0.8f) atomicAdd(&h_att[t], 1u);
        }
        if (te == 1) pn0++;
        else if (te == 0) pn1++;
    }
    atomicAdd(&h_pn[0], pn0);
    atomicAdd(&h_pn[1], pn1);
    __syncthreads();

    if (tid < NCLS) {
        atomicAdd(&ws_u[OFF_CNT_SEG + n * NCLS + tid], h_seg[tid]);
        atomicAdd(&ws_u[OFF_CNT_ATT + n * NCLS + tid], h_att[tid]);
    }
    if (tid == 0) {
        atomicAdd(&ws_u[OFF_POSNEG + 0], h_pn[0]);
        atomicAdd(&ws_u[OFF_POSNEG + 1], h_pn[1]);
    }
}

// ---------------------------------------------------------------------------
// Tiny kernel: histogram -> per-image per-class weights
// w = (cnt>0) * UPPER*(1 - cnt/total) + 1
// ---------------------------------------------------------------------------
__global__ void weights_kernel(const unsigned* __restrict__ ws_u,
                               float* __restrict__ ws_f) {
    const int tid = threadIdx.x;
    if (tid >= NIMG * NCLS) return;
    const int n = tid / NCLS;

    float tot = 0.f;
    for (int c = 0; c < NCLS; ++c) tot += (float)ws_u[OFF_CNT_SEG + n * NCLS + c];
    float den = fmaxf(tot, 1.0f);
    unsigned cnt = ws_u[OFF_CNT_SEG + tid];
    ws_f[OFF_W_SEG + tid] = (cnt > 0u) ? ((1.0f - (float)cnt / den) + 1.0f) : 1.0f;

    float tot2 = 0.f;
    for (int c = 0; c < NCLS; ++c) tot2 += (float)ws_u[OFF_CNT_ATT + n * NCLS + c];
    float den2 = fmaxf(tot2, 1.0f);
    unsigned cnt2 = ws_u[OFF_CNT_ATT + tid];
    ws_f[OFF_W_ATT + tid] = (cnt2 > 0u) ? ((1.0f - (float)cnt2 / den2) + 1.0f) : 1.0f;
}

// ---------------------------------------------------------------------------
// Pass B: single stream over segin — log-softmax, weighted NLL (seg + att),
// balanced BCE. Per-block partial sums written to workspace (deterministic).
// ---------------------------------------------------------------------------
__global__ void loss_kernel(const float* __restrict__ segin,
                            const float* __restrict__ edgein,
                            const int* __restrict__ segmask,
                            const int* __restrict__ edgemask,
                            const unsigned* __restrict__ ws_u,
                            const float* __restrict__ ws_f,
                            float* __restrict__ partials) {
    __shared__ float w_seg_s[NCLS];
    __shared__ float w_att_s[NCLS];
    __shared__ float red[BLK];
    const int tid = threadIdx.x;
    const int n = blockIdx.y;
    if (tid < NCLS) {
        w_seg_s[tid] = ws_f[OFF_W_SEG + n * NCLS + tid];
        w_att_s[tid] = ws_f[OFF_W_ATT + n * NCLS + tid];
    }
    __syncthreads();

    const float pos = (float)ws_u[OFF_POSNEG + 0];
    const float neg = (float)ws_u[OFF_POSNEG + 1];
    const float s = fmaxf(pos + neg, 1.0f);
    const float rp = pos / s, rn = neg / s;

    float aA = 0.f, aB = 0.f, bA = 0.f, bB = 0.f, eE = 0.f;
    const int base = blockIdx.x * CHUNK;
    const float* img = segin + (size_t)n * NCLS * HW;

    for (int i = 0; i < CHUNK / BLK; ++i) {
        const int p = base + i * BLK + tid;

        float x[NCLS];
#pragma unroll
        for (int c = 0; c < NCLS; ++c) x[c] = img[(size_t)c * HW + p];

        float m = x[0];
#pragma unroll
        for (int c = 1; c < NCLS; ++c) m = fmaxf(m, x[c]);
        float ssum = 0.f;
#pragma unroll
        for (int c = 0; c < NCLS; ++c) ssum += __expf(x[c] - m);
        const float lse = m + __logf(ssum);

        const int gp = n * HW + p;
        const int t = segmask[gp];
        const bool v = (t >= 0) && (t < NCLS);
        const int tc = v ? t : 0;
        float xt = x[0];
#pragma unroll
        for (int c = 1; c < NCLS; ++c) xt = (tc == c) ? x[c] : xt;
        const float lp = xt - lse;

        const float e = edgein[gp];
        if (v) {
            const float w1 = w_seg_s[tc];
            aA += w1 * lp; aB += w1;
            if (e > 0.8f) {
                const float w2 = w_att_s[tc];
                bA += w2 * lp; bB += w2;
            }
        }

        // balanced BCE-with-logits on edge channel
        const int te = edgemask[gp];
        const float wb = (te == 1) ? rn : ((te == 0) ? rp : 0.f);
        const float tf = (float)te;
        const float bce = fmaxf(e, 0.f) - e * tf + log1pf(__expf(-fabsf(e)));
        eE += wb * bce;
    }

    float q[5] = {aA, aB, bA, bB, eE};
    for (int k = 0; k < 5; ++k) {
        red[tid] = q[k];
        __syncthreads();
        for (int st = BLK / 2; st > 0; st >>= 1) {
            if (tid < st) red[tid] += red[tid + st];
            __syncthreads();
        }
        if (tid == 0)
            partials[(size_t)(n * BPI + blockIdx.x) * 8 + k] = red[0];
        __syncthreads();
    }
}

// ---------------------------------------------------------------------------
// Final reduction: one wave. Sum 64 f32 block-partials per (image, quantity)
// with a single V_WMMA_F32_16X16X4_F32 (A = ones). With A==1, every row of D
// equals the 16 column sums of B, so sum(d[0] over all 32 lanes) == 2*total —
// correct regardless of the exact B operand lane striping.
// ---------------------------------------------------------------------------
__device__ __forceinline__ float wave_sum64_wmma(float v0, float v1) {
    v2f a; a[0] = 1.0f; a[1] = 1.0f;
    v2f b; b[0] = v0;   b[1] = v1;
    v8f c = {};
    v8f d = __builtin_amdgcn_wmma_f32_16x16x4_f32(
        false, a, false, b, (short)0, c, false, false);
    float cs = d[0];
#pragma unroll
    for (int msk = 1; msk < 32; msk <<= 1) cs += __shfl_xor(cs, msk, 32);
    return 0.5f * cs;
}

__global__ void final_kernel(const float* __restrict__ partials,
                             float* __restrict__ out) {
    __shared__ float tot[NIMG][5];
    const int lane = threadIdx.x;  // 32 threads, full wave (EXEC all ones)
    for (int n = 0; n < NIMG; ++n) {
        for (int q = 0; q < 5; ++q) {
            const float* pb = partials + (size_t)(n * BPI) * 8 + q;
            const float v0 = pb[(size_t)(lane * 2 + 0) * 8];
            const float v1 = pb[(size_t)(lane * 2 + 1) * 8];
            const float t = wave_sum64_wmma(v0, v1);
            if (lane == 0) tot[n][q] = t;
        }
    }
    __syncthreads();
    if (lane == 0) {
        float seg = 0.f, att = 0.f, E = 0.f;
        for (int n = 0; n < NIMG; ++n) {
            seg += -tot[n][0] / fmaxf(tot[n][1], 1e-12f);
            att += -tot[n][2] / fmaxf(tot[n][3], 1e-12f);
            E += tot[n][4];
        }
        out[0] = seg;                                  // seg_loss
        out[1] = 20.0f * E / (float)(NIMG * HW);       // edge_loss
        out[2] = att;                                  // att_loss
        out[3] = 0.0f;                                 // dual_loss
    }
}

// ---------------------------------------------------------------------------
extern "C" void kernel_launch(void* const* d_in, const int* in_sizes, int n_in,
                              void* d_out, int out_size, void* d_ws, size_t ws_size,
                              hipStream_t stream) {
    const float* segin   = (const float*)d_in[0];
    const float* edgein  = (const float*)d_in[1];
    const int*   segmask = (const int*)d_in[2];
    const int*   edgemask= (const int*)d_in[3];
    float* out = (float*)d_out;

    unsigned* ws_u = (unsigned*)d_ws;
    float*    ws_f = (float*)d_ws;
    float*    partials = ws_f + OFF_PART;

    // zero the integer-counter region (first 4 KB) — graph-capturable memset
    hipMemsetAsync(d_ws, 0, 4096, stream);

    dim3 grid(BPI, NIMG);
    histo_kernel<<<grid, BLK, 0, stream>>>(edgein, segmask, edgemask, ws_u);
    weights_kernel<<<1, BLK, 0, stream>>>(ws_u, ws_f);
    loss_kernel<<<grid, BLK, 0, stream>>>(segin, edgein, segmask, edgemask,
                                          ws_u, ws_f, partials);
    final_kernel<<<1, 32, 0, stream>>>(partials, out);
}